// RenderMesh_2551210574268
// MI455X (gfx1250) — compile-verified
//
#include <hip/hip_runtime.h>
#include <hip/hip_bf16.h>
#include <math.h>

// ---------------- problem constants ----------------
#define BB    2
#define VV    5023
#define FF    9976
#define HH    256
#define PP    (HH * HH)          // 65536 pixels per image
#define FPAD  9984               // ceil(FF/16)*16
#define NTILE (FPAD / 16)        // 624 face tiles
#define EPSZ  1e-5f

typedef float v2f __attribute__((ext_vector_type(2)));
typedef float v8f __attribute__((ext_vector_type(8)));

// ---------------- workspace layout (float offsets) ----------------
#define OFF_VNDC  0                              // B*V*3 -> 30138, round 30144
#define OFF_CW0   30144                          // B*FPAD*4
#define OFF_CW1   (OFF_CW0 + BB * FPAD * 4)
#define OFF_CZ    (OFF_CW1 + BB * FPAD * 4)
#define OFF_IZ    (OFF_CZ  + BB * FPAD * 4)      // per-face {1/z0,1/z1,1/z2,0}
#define OFF_VN    (OFF_IZ  + BB * FPAD * 4)      // B*V*3 -> 30138, round 30144
#define OFF_RAST  (OFF_VN  + 30144)              // B*P*2 (float2 {iz, face} per pixel)

__device__ __forceinline__ float axcoord(int i) {
    return 1.0f - (2.0f * (float)i + 1.0f) / (float)HH;
}

// ---------------- 1) vertex transform to NDC ----------------
__global__ void k_transform(const float* __restrict__ verts,
                            const float* __restrict__ tm,
                            const float* __restrict__ focal,
                            float* __restrict__ vndc) {
    int idx = blockIdx.x * blockDim.x + threadIdx.x;
    if (idx >= BB * VV) return;
    int b = idx / VV;
    const float* M = tm + b * 12;          // (3,4): R = M[r*4+c], T = M[r*4+3]
    float vx = verts[idx * 3 + 0], vy = verts[idx * 3 + 1], vz = verts[idx * 3 + 2];
    // cam_j = sum_i v_i * R[i][j] + T_j
    float cx = vx * M[0] + vy * M[4] + vz * M[8]  + M[3];
    float cy = vx * M[1] + vy * M[5] + vz * M[9]  + M[7];
    float cz = vx * M[2] + vy * M[6] + vz * M[10] + M[11];
    float f = focal[b];
    vndc[idx * 3 + 0] = f * cx / cz;
    vndc[idx * 3 + 1] = f * cy / cz;
    vndc[idx * 3 + 2] = cz;
}

// ---------------- 2) per-face edge-function setup ----------------
__global__ void k_setup(const float* __restrict__ vndc,
                        const int*   __restrict__ faces,
                        float* __restrict__ cw0,
                        float* __restrict__ cw1,
                        float* __restrict__ cz,
                        float* __restrict__ izd) {
    int idx = blockIdx.x * blockDim.x + threadIdx.x;
    if (idx >= BB * FPAD) return;
    int b = idx / FPAD, fp = idx % FPAD;

    float A0 = 0.f, B0 = 0.f, C0 = -1.f;      // invalid faces: w0 == -1, never inside
    float A1 = 0.f, B1 = 0.f, C1 = -1.f;
    float AZ = 0.f, BZ = 0.f, CZ = 0.f;
    float iz0 = 1.f, iz1 = 1.f, iz2 = 1.f;

    if (fp < FF) {
        int i0 = faces[fp * 3 + 0], i1 = faces[fp * 3 + 1], i2 = faces[fp * 3 + 2];
        const float* p0 = vndc + ((size_t)b * VV + i0) * 3;
        const float* p1 = vndc + ((size_t)b * VV + i1) * 3;
        const float* p2 = vndc + ((size_t)b * VV + i2) * 3;
        float x0 = p0[0], y0 = p0[1], z0 = p0[2];
        float x1 = p1[0], y1 = p1[1], z1 = p1[2];
        float x2 = p2[0], y2 = p2[1], z2 = p2[2];
        float den = (y1 - y2) * (x0 - x2) + (x2 - x1) * (y0 - y2);
        bool ok = fabsf(den) > 1e-8f;
        float dens = ok ? den : 1.0f;
        float z0s = (z0 > EPSZ) ? z0 : 1.0f;
        float z1s = (z1 > EPSZ) ? z1 : 1.0f;
        float z2s = (z2 > EPSZ) ? z2 : 1.0f;
        iz0 = 1.0f / z0s; iz1 = 1.0f / z1s; iz2 = 1.0f / z2s;
        bool valid = ok && (z0 > EPSZ) && (z1 > EPSZ) && (z2 > EPSZ);
        if (valid) {
            float inv = 1.0f / dens;
            A0 = (y1 - y2) * inv;
            B0 = (x2 - x1) * inv;
            C0 = -((y1 - y2) * x2 + (x2 - x1) * y2) * inv;
            A1 = (y2 - y0) * inv;
            B1 = (x0 - x2) * inv;
            C1 = -((y2 - y0) * x2 + (x0 - x2) * y2) * inv;
            float d02 = iz0 - iz2, d12 = iz1 - iz2;
            AZ = A0 * d02 + A1 * d12;
            BZ = B0 * d02 + B1 * d12;
            CZ = C0 * d02 + C1 * d12 + iz2;
        }
    }
    size_t o = (size_t)idx * 4;
    cw0[o + 0] = A0; cw0[o + 1] = B0; cw0[o + 2] = C0; cw0[o + 3] = 0.f;
    cw1[o + 0] = A1; cw1[o + 1] = B1; cw1[o + 2] = C1; cw1[o + 3] = 0.f;
    cz [o + 0] = AZ; cz [o + 1] = BZ; cz [o + 2] = CZ; cz [o + 3] = 0.f;
    izd[o + 0] = iz0; izd[o + 1] = iz1; izd[o + 2] = iz2; izd[o + 3] = 0.f;
}

// ---------------- 3) WMMA rasterizer ----------------
// One wave per 32-pixel run (two 16-px B-fragments share each A-fragment load).
// Per face tile: 3 x global_load_b64 feed 6 x v_wmma_f32_16x16x4_f32.
__global__ void k_raster(const float* __restrict__ cw0,
                         const float* __restrict__ cw1,
                         const float* __restrict__ cz,
                         float2* __restrict__ rast) {
    int wave = blockIdx.x * (blockDim.x >> 5) + (threadIdx.x >> 5);
    const int runsPerImg = PP / 32;             // 2048
    int b  = wave / runsPerImg;
    int pr = wave % runsPerImg;
    int lane = threadIdx.x & 31;
    int n    = lane & 15;                       // pixel column N within a 16-px group
    int half = lane >> 4;                       // 0: K=0/1 rows, 1: K=2/3 rows

    int pA = pr * 32 + n;
    int pB = pA + 16;
    float pxA = axcoord(pA & (HH - 1)), pyA = axcoord(pA >> 8);
    float pxB = axcoord(pB & (HH - 1)), pyB = axcoord(pB >> 8);

    // B fragment (4x16): row0 = px, row1 = py, row2 = 1, row3 = 0
    v2f bfA, bfB;
    bfA.x = half ? 1.0f : pxA;  bfA.y = half ? 0.0f : pyA;
    bfB.x = half ? 1.0f : pxB;  bfB.y = half ? 0.0f : pyB;

    const float* base0 = cw0 + (size_t)b * FPAD * 4;
    const float* base1 = cw1 + (size_t)b * FPAD * 4;
    const float* basez = cz  + (size_t)b * FPAD * 4;
    const int aoff = half * 2;

    v8f zacc = {};
    float bestIzA = 1e-8f, bestIzB = 1e-8f;     // subsumes the iz > 1e-8 test
    int bfaceA = -1, bfaceB = -1;

    for (int t = 0; t < NTILE; ++t) {
        size_t fo = ((size_t)(t * 16 + n)) * 4 + aoff;
        v2f a0 = *(const v2f*)(base0 + fo);
        v2f a1 = *(const v2f*)(base1 + fo);
        v2f az = *(const v2f*)(basez + fo);
        __builtin_prefetch(base0 + fo + 64, 0, 3);   // next tile -> near cache
        __builtin_prefetch(base1 + fo + 64, 0, 3);
        __builtin_prefetch(basez + fo + 64, 0, 3);

        v8f c0A = __builtin_amdgcn_wmma_f32_16x16x4_f32(false, a0, false, bfA,
                                                        (short)0, zacc, false, false);
        v8f c1A = __builtin_amdgcn_wmma_f32_16x16x4_f32(false, a1, false, bfA,
                                                        (short)0, zacc, false, false);
        v8f c2A = __builtin_amdgcn_wmma_f32_16x16x4_f32(false, az, false, bfA,
                                                        (short)0, zacc, false, false);
        v8f c0B = __builtin_amdgcn_wmma_f32_16x16x4_f32(false, a0, false, bfB,
                                                        (short)0, zacc, false, false);
        v8f c1B = __builtin_amdgcn_wmma_f32_16x16x4_f32(false, a1, false, bfB,
                                                        (short)0, zacc, false, false);
        v8f c2B = __builtin_amdgcn_wmma_f32_16x16x4_f32(false, az, false, bfB,
                                                        (short)0, zacc, false, false);
#pragma unroll
        for (int m = 0; m < 8; ++m) {
            int fidx = t * 16 + half * 8 + m;
            {
                float w0 = c0A[m], w1 = c1A[m], izv = c2A[m];
                bool inside = (w0 >= 0.0f) && (w1 >= 0.0f) && ((w0 + w1) <= 1.0f);
                bool take   = inside && (izv > bestIzA);  // strict: earlier face wins ties
                bestIzA = take ? izv  : bestIzA;
                bfaceA  = take ? fidx : bfaceA;
            }
            {
                float w0 = c0B[m], w1 = c1B[m], izv = c2B[m];
                bool inside = (w0 >= 0.0f) && (w1 >= 0.0f) && ((w0 + w1) <= 1.0f);
                bool take   = inside && (izv > bestIzB);
                bestIzB = take ? izv  : bestIzB;
                bfaceB  = take ? fidx : bfaceB;
            }
        }
    }

    // combine the two half-lanes covering the same pixel (faces m and m+8)
    {
        float oIz = __shfl_xor(bestIzA, 16, 32);
        int   oF  = __shfl_xor(bfaceA,  16, 32);
        bool take = (oIz > bestIzA) ||
                    ((oIz == bestIzA) && ((unsigned)oF < (unsigned)bfaceA));
        bestIzA = take ? oIz : bestIzA;
        bfaceA  = take ? oF  : bfaceA;
    }
    {
        float oIz = __shfl_xor(bestIzB, 16, 32);
        int   oF  = __shfl_xor(bfaceB,  16, 32);
        bool take = (oIz > bestIzB) ||
                    ((oIz == bestIzB) && ((unsigned)oF < (unsigned)bfaceB));
        bestIzB = take ? oIz : bestIzB;
        bfaceB  = take ? oF  : bfaceB;
    }

    if (half == 0) {
        rast[(size_t)b * PP + pA] = make_float2(bestIzA, __int_as_float(bfaceA));
        rast[(size_t)b * PP + pB] = make_float2(bestIzB, __int_as_float(bfaceB));
    }
}

// ---------------- 4) vertex normals ----------------
__global__ void k_zero(float* __restrict__ vn, int nelems) {
    int idx = blockIdx.x * blockDim.x + threadIdx.x;
    if (idx < nelems) vn[idx] = 0.0f;
}

__global__ void k_face_normals(const float* __restrict__ verts,
                               const int*   __restrict__ faces,
                               float* __restrict__ vn) {
    int idx = blockIdx.x * blockDim.x + threadIdx.x;
    if (idx >= BB * FF) return;
    int b = idx / FF, f = idx % FF;
    int i0 = faces[f * 3 + 0], i1 = faces[f * 3 + 1], i2 = faces[f * 3 + 2];
    const float* p0 = verts + ((size_t)b * VV + i0) * 3;
    const float* p1 = verts + ((size_t)b * VV + i1) * 3;
    const float* p2 = verts + ((size_t)b * VV + i2) * 3;
    float e1x = p1[0] - p0[0], e1y = p1[1] - p0[1], e1z = p1[2] - p0[2];
    float e2x = p2[0] - p0[0], e2y = p2[1] - p0[1], e2z = p2[2] - p0[2];
    float nx = e1y * e2z - e1z * e2y;
    float ny = e1z * e2x - e1x * e2z;
    float nz = e1x * e2y - e1y * e2x;
    float* v0 = vn + ((size_t)b * VV + i0) * 3;
    float* v1 = vn + ((size_t)b * VV + i1) * 3;
    float* v2 = vn + ((size_t)b * VV + i2) * 3;
    atomicAdd(v0 + 0, nx); atomicAdd(v0 + 1, ny); atomicAdd(v0 + 2, nz);
    atomicAdd(v1 + 0, nx); atomicAdd(v1 + 1, ny); atomicAdd(v1 + 2, nz);
    atomicAdd(v2 + 0, nx); atomicAdd(v2 + 1, ny); atomicAdd(v2 + 2, nz);
}

// ---------------- 5) deferred shading ----------------
__device__ __forceinline__ void norm3(float& x, float& y, float& z) {
    float n = sqrtf(x * x + y * y + z * z);
    float inv = 1.0f / fmaxf(n, 1e-6f);
    x *= inv; y *= inv; z *= inv;
}

__global__ void k_shade(const float2* __restrict__ rast,
                        const float*  __restrict__ cw0,
                        const float*  __restrict__ cw1,
                        const float*  __restrict__ izd,
                        const int*    __restrict__ faces,
                        const float*  __restrict__ verts,
                        const float*  __restrict__ vn,
                        const float*  __restrict__ tm,
                        float* __restrict__ out) {
    int idx = blockIdx.x * blockDim.x + threadIdx.x;
    if (idx >= BB * PP) return;
    int b = idx / PP, p = idx % PP;

    float2 r = rast[idx];
    int face = __float_as_int(r.y);

    float rr = 1.0f, gg = 1.0f, bb = 1.0f;   // background
    float alpha = 0.0f;

    if (face >= 0) {
        alpha = 1.0f;
        float px = axcoord(p & (HH - 1)), py = axcoord(p >> 8);
        // recompute raw barycentrics of the winning face from stored coefficients
        const float* c0p = cw0 + ((size_t)b * FPAD + face) * 4;
        const float* c1p = cw1 + ((size_t)b * FPAD + face) * 4;
        float w0 = c0p[0] * px + c0p[1] * py + c0p[2];
        float w1 = c1p[0] * px + c1p[1] * py + c1p[2];
        float zsafe = 1.0f / r.x;
        const float* izp = izd + ((size_t)b * FPAD + face) * 4;
        float w0c = w0 * izp[0] * zsafe;
        float w1c = w1 * izp[1] * zsafe;
        float w2c = (1.0f - w0 - w1) * izp[2] * zsafe;

        int i0 = faces[face * 3 + 0], i1 = faces[face * 3 + 1], i2 = faces[face * 3 + 2];
        const float* g0 = verts + ((size_t)b * VV + i0) * 3;
        const float* g1 = verts + ((size_t)b * VV + i1) * 3;
        const float* g2 = verts + ((size_t)b * VV + i2) * 3;
        const float* m0 = vn + ((size_t)b * VV + i0) * 3;
        const float* m1 = vn + ((size_t)b * VV + i1) * 3;
        const float* m2 = vn + ((size_t)b * VV + i2) * 3;

        float n0x = m0[0], n0y = m0[1], n0z = m0[2]; norm3(n0x, n0y, n0z);
        float n1x = m1[0], n1y = m1[1], n1z = m1[2]; norm3(n1x, n1y, n1z);
        float n2x = m2[0], n2y = m2[1], n2z = m2[2]; norm3(n2x, n2y, n2z);

        float posx = w0c * g0[0] + w1c * g1[0] + w2c * g2[0];
        float posy = w0c * g0[1] + w1c * g1[1] + w2c * g2[1];
        float posz = w0c * g0[2] + w1c * g1[2] + w2c * g2[2];

        float nx = w0c * n0x + w1c * n1x + w2c * n2x;
        float ny = w0c * n0y + w1c * n1y + w2c * n2y;
        float nz = w0c * n0z + w1c * n1z + w2c * n2z;
        norm3(nx, ny, nz);

        // cam_pos[k] = -sum_j T[j] * R[k][j]
        const float* M = tm + b * 12;
        float Tx = M[3], Ty = M[7], Tz = M[11];
        float cpx = -(Tx * M[0] + Ty * M[1] + Tz * M[2]);
        float cpy = -(Tx * M[4] + Ty * M[5] + Tz * M[6]);
        float cpz = -(Tx * M[8] + Ty * M[9] + Tz * M[10]);

        float lx = 0.0f - posx, ly = 1.0f - posy, lz = 3.0f - posz;
        norm3(lx, ly, lz);
        float vx = cpx - posx, vy = cpy - posy, vz = cpz - posz;
        norm3(vx, vy, vz);

        float cosNL = nx * lx + ny * ly + nz * lz;
        float diffuse = 0.3f * fmaxf(cosNL, 0.0f);
        float rx = 2.0f * cosNL * nx - lx;
        float ry = 2.0f * cosNL * ny - ly;
        float rz = 2.0f * cosNL * nz - lz;
        float s = fmaxf(vx * rx + vy * ry + vz * rz, 0.0f);
        float s2 = s * s, s4 = s2 * s2, s8 = s4 * s4;
        float spec = 0.12f * (s8 * s2);       // s^10

        float shade = 0.5f + diffuse;
        rr = (142.0f / 255.0f) * shade + spec;
        gg = (179.0f / 255.0f) * shade + spec;
        bb = (247.0f / 255.0f) * shade + spec;
    }

    // images: (B,3,H,H) * 255, then alpha: (B,1,H,H)
    size_t ib = (size_t)b * 3 * PP;
    out[ib + 0 * PP + p] = rr * 255.0f;
    out[ib + 1 * PP + p] = gg * 255.0f;
    out[ib + 2 * PP + p] = bb * 255.0f;
    out[(size_t)BB * 3 * PP + (size_t)b * PP + p] = alpha;
}

// ---------------- launch ----------------
extern "C" void kernel_launch(void* const* d_in, const int* in_sizes, int n_in,
                              void* d_out, int out_size, void* d_ws, size_t ws_size,
                              hipStream_t stream) {
    const float* vertices = (const float*)d_in[0];   // (B,V,3)
    const float* tm       = (const float*)d_in[1];   // (B,3,4)
    const float* focal    = (const float*)d_in[2];   // (B,1)
    const int*   faces    = (const int*)d_in[3];     // (F,3)

    float* ws   = (float*)d_ws;
    float* vndc = ws + OFF_VNDC;
    float* cw0  = ws + OFF_CW0;
    float* cw1  = ws + OFF_CW1;
    float* cz   = ws + OFF_CZ;
    float* izd  = ws + OFF_IZ;
    float* vn   = ws + OFF_VN;
    float2* rast = (float2*)(ws + OFF_RAST);
    float* out  = (float*)d_out;

    k_transform<<<(BB * VV + 255) / 256, 256, 0, stream>>>(vertices, tm, focal, vndc);
    k_setup<<<(BB * FPAD) / 256, 256, 0, stream>>>(vndc, faces, cw0, cw1, cz, izd);
    k_zero<<<(BB * VV * 3 + 255) / 256, 256, 0, stream>>>(vn, BB * VV * 3);
    k_face_normals<<<(BB * FF + 255) / 256, 256, 0, stream>>>(vertices, faces, vn);
    // one wave (32 lanes) per 32-pixel run -> B*P/32 waves, 8 waves per block
    k_raster<<<(BB * PP / 32) / 8, 256, 0, stream>>>(cw0, cw1, cz, rast);
    k_shade<<<(BB * PP) / 256, 256, 0, stream>>>(rast, cw0, cw1, izd, faces,
                                                 vertices, vn, tm, out);
}